// GIN_30932354466309
// MI455X (gfx1250) — compile-verified
//
#include <hip/hip_runtime.h>
#include <hip/hip_bf16.h>

typedef float v2f __attribute__((ext_vector_type(2)));
typedef float v8f __attribute__((ext_vector_type(8)));

#define NNODES 65536   // G*N = 128*512
#define FEAT   128
#define NGRAPH 128
#define NPG    512
#define NCLASS 10
#define NLAYER 5

// ---------------------------------------------------------------------------
// Scatter-add: pooled[dst[e]][:] += h[src[e]][:]
// One wave32 per edge; each lane handles 4 consecutive floats (float4 load +
// 4 global_atomic_add_f32). readfirstlane scalarizes the edge index so the
// endpoint fetches become s_load instead of 32 redundant vector loads.
// Edge lists + both feature buffers are L2-resident (192 MB).
// ---------------------------------------------------------------------------
__global__ __launch_bounds__(256)
void gin_scatter_add(const float* hin, const int* __restrict__ src,
                     const int* __restrict__ dst, float* pooled, int nedges) {
    unsigned long long tid = (unsigned long long)blockIdx.x * 256ull + threadIdx.x;
    int e    = (int)(tid >> 5);
    int lane = (int)(tid & 31);
    if (e >= nedges) return;
    e = __builtin_amdgcn_readfirstlane(e);   // uniform within wave -> SGPR
    int s = src[e];
    int d = dst[e];
    const float4 v = *((const float4*)(hin + (size_t)s * FEAT) + lane);
    float* dp = pooled + (size_t)d * FEAT + lane * 4;
    atomicAdd(dp + 0, v.x);
    atomicAdd(dp + 1, v.y);
    atomicAdd(dp + 2, v.z);
    atomicAdd(dp + 3, v.w);
}

// ---------------------------------------------------------------------------
// Fused GEMM: out = relu((s*hin [+ pooled]) @ Wmat + bias), exact fp32 via
// V_WMMA_F32_16X16X4_F32. Block = 256 threads = 8 waves; W staged in 64KB LDS
// with k-pairs interleaved: ldsW[(k>>1)*256 + n*2 + (k&1)] = W[k][n], so a
// B fragment (W[ka][n], W[ka+1][n]) is one aligned ds_load_b64.
// Wave w owns rows [ (blockIdx*8+w)*16 , +16 ) and all 8 column tiles.
// A frag (lane = M, half = lane>>4): v{0,1} = A[M][k0 + 2*half + {0,1}]
// B frag (lane&15 = N):              v{0,1} = W[k0 + 2*half + {0,1}][N]
// D frag:                            v[r]   = D[r + 8*half][N]
// In-place (out==hin) is safe: row m's output depends only on row m's input,
// and all loads precede all stores within the owning wave.
// ---------------------------------------------------------------------------
template <bool HAS_POOLED>
__global__ __launch_bounds__(256)
void gin_gemm_relu(const float* hin, const float* pooled,
                   const float* __restrict__ Wmat, const float* __restrict__ bias,
                   const float* eps, int layer, float* out) {
    __shared__ float ldsW[FEAT * FEAT];   // 64 KB, k-pair interleaved
    const int t = threadIdx.x;
    #pragma unroll 4
    for (int i = t; i < FEAT * FEAT; i += 256) {
        int k = i >> 7;          // row (K index)
        int n = i & 127;         // col (N index)
        ldsW[(k >> 1) * 256 + n * 2 + (k & 1)] = Wmat[i];
    }
    __syncthreads();

    const float s    = HAS_POOLED ? (1.0f + eps[layer]) : 1.0f;
    const int wave   = t >> 5;
    const int lane   = t & 31;
    const int half   = lane >> 4;
    const int li     = lane & 15;
    const int row0   = (blockIdx.x * 8 + wave) * 16;

    v8f acc[8];
    #pragma unroll
    for (int nt = 0; nt < 8; ++nt)
        #pragma unroll
        for (int r = 0; r < 8; ++r) acc[nt][r] = 0.0f;

    const size_t arow = (size_t)(row0 + li) * FEAT;
    #pragma unroll 4
    for (int k0 = 0; k0 < FEAT; k0 += 4) {
        const int ka = k0 + 2 * half;              // even -> 8B aligned
        v2f a = *(const v2f*)(hin + arow + ka);
        if (HAS_POOLED) {
            v2f p2 = *(const v2f*)(pooled + arow + ka);
            a.x = fmaf(s, a.x, p2.x);
            a.y = fmaf(s, a.y, p2.y);
        }
        const float* brow = &ldsW[(ka >> 1) * 256 + li * 2];
        #pragma unroll
        for (int nt = 0; nt < 8; ++nt) {
            v2f bf = *(const v2f*)(brow + nt * 32);   // one ds_load_b64
            acc[nt] = __builtin_amdgcn_wmma_f32_16x16x4_f32(
                false, a, false, bf, (short)0, acc[nt], false, false);
        }
    }

    #pragma unroll
    for (int nt = 0; nt < 8; ++nt) {
        const float bv = bias[nt * 16 + li];
        #pragma unroll
        for (int r = 0; r < 8; ++r) {
            float v = acc[nt][r] + bv;
            v = v > 0.0f ? v : 0.0f;
            out[(size_t)(row0 + r + 8 * half) * FEAT + nt * 16 + li] = v;
        }
    }
}

// ---------------------------------------------------------------------------
// Readout: hg[g][f] = sum_n H[g*512+n][f]  (coalesced across f)
// ---------------------------------------------------------------------------
__global__ __launch_bounds__(128)
void gin_readout(const float* __restrict__ H, float* __restrict__ hg) {
    const int g = blockIdx.x;
    const int f = threadIdx.x;
    const float* base = H + (size_t)g * NPG * FEAT + f;
    float sum = 0.0f;
    #pragma unroll 8
    for (int n = 0; n < NPG; ++n) sum += base[(size_t)n * FEAT];
    hg[g * FEAT + f] = sum;
}

// ---------------------------------------------------------------------------
// FC2 + softmax: out[g][:] = softmax(hg2[g] @ fc2_w + fc2_b). 128x10 — trivial.
// ---------------------------------------------------------------------------
__global__ __launch_bounds__(128)
void gin_fc2_softmax(const float* __restrict__ hg2, const float* __restrict__ w,
                     const float* __restrict__ bias, float* __restrict__ out) {
    const int g = blockIdx.x * blockDim.x + threadIdx.x;
    if (g >= NGRAPH) return;
    const float* row = hg2 + (size_t)g * FEAT;
    float logits[NCLASS];
    float mx = -3.402823466e38f;
    #pragma unroll
    for (int c = 0; c < NCLASS; ++c) {
        float acc = bias[c];
        for (int k = 0; k < FEAT; ++k) acc += row[k] * w[k * NCLASS + c];
        logits[c] = acc;
        mx = fmaxf(mx, acc);
    }
    float sum = 0.0f;
    #pragma unroll
    for (int c = 0; c < NCLASS; ++c) { logits[c] = __expf(logits[c] - mx); sum += logits[c]; }
    const float inv = 1.0f / sum;
    #pragma unroll
    for (int c = 0; c < NCLASS; ++c) out[(size_t)g * NCLASS + c] = logits[c] * inv;
}

// ---------------------------------------------------------------------------
extern "C" void kernel_launch(void* const* d_in, const int* in_sizes, int n_in,
                              void* d_out, int out_size, void* d_ws, size_t ws_size,
                              hipStream_t stream) {
    const float* x     = (const float*)d_in[0];
    const float* eps   = (const float*)d_in[1];
    const float* W     = (const float*)d_in[2];
    const float* b     = (const float*)d_in[3];
    const float* fc1_w = (const float*)d_in[4];
    const float* fc1_b = (const float*)d_in[5];
    const float* fc2_w = (const float*)d_in[6];
    const float* fc2_b = (const float*)d_in[7];
    const int*   src   = (const int*)d_in[8];
    const int*   dst   = (const int*)d_in[9];
    const int nedges   = in_sizes[8];

    const size_t featBytes = (size_t)NNODES * FEAT * sizeof(float);  // 32 MB
    char* ws = (char*)d_ws;
    float* H   = (float*)ws;                         // node features (in-place)
    float* P   = (float*)(ws + featBytes);           // pooled aggregation
    float* hg  = (float*)(ws + 2 * featBytes);       // graph readout 128x128
    float* hg2 = hg + NGRAPH * FEAT;                 // fc1 output    128x128

    const int scatterBlocks = (nedges + 7) / 8;      // 8 edges per 256-thread block
    const int gemmBlocks    = NNODES / 128;          // 8 waves x 16 rows per block

    for (int layer = 0; layer < NLAYER; ++layer) {
        const float* hin = (layer == 0) ? x : H;
        hipMemsetAsync(P, 0, featBytes, stream);
        gin_scatter_add<<<scatterBlocks, 256, 0, stream>>>(hin, src, dst, P, nedges);
        gin_gemm_relu<true><<<gemmBlocks, 256, 0, stream>>>(
            hin, P, W + (size_t)layer * FEAT * FEAT, b + (size_t)layer * FEAT,
            eps, layer, H);
    }

    gin_readout<<<NGRAPH, 128, 0, stream>>>(H, hg);
    // FC1: 128 rows -> single 8-wave block; no pooled term, scale = 1
    gin_gemm_relu<false><<<1, 256, 0, stream>>>(hg, nullptr, fc1_w, fc1_b,
                                                nullptr, 0, hg2);
    gin_fc2_softmax<<<1, 128, 0, stream>>>(hg2, fc2_w, fc2_b, (float*)d_out);
}